// Net_343597383912
// MI455X (gfx1250) — compile-verified
//
#include <hip/hip_runtime.h>
#include <hip/hip_bf16.h>

// ---------------------------------------------------------------------------
// Sizes fixed by the reference
// ---------------------------------------------------------------------------
#define N_TOK     4096
#define VEC_DIM   300
#define DIM       128
#define EDGES     2000

typedef float v2f __attribute__((ext_vector_type(2)));
typedef float v8f __attribute__((ext_vector_type(8)));

// ---------------------------------------------------------------------------
// Kernel 1: emb[i,:] = bias + data_vecs[data[i],:] @ W   (4096x300 * 300x128)
// One wave32 per 16x16 output tile, native f32 WMMA 16x16x4, K = 75 steps.
// ---------------------------------------------------------------------------
__global__ __launch_bounds__(128) void emb_gemm_wmma(
    const int*   __restrict__ data,    // [4096]
    const float* __restrict__ vecs,    // [VOCAB, 300]
    const float* __restrict__ W,       // [300, 128]
    const float* __restrict__ bias,    // [128]
    float*       __restrict__ emb)     // [4096, 128]
{
    const int wave  = (blockIdx.x * blockDim.x + threadIdx.x) >> 5; // 0..2047
    const int lane  = threadIdx.x & 31;
    const int tileM = wave >> 3;            // 0..255  (4096/16)
    const int tileN = wave & 7;             // 0..7    (128/16)
    const int row0  = tileM * 16;
    const int col0  = tileN * 16;
    const int half  = lane >> 4;            // 0 or 1
    const int l16   = lane & 15;

    // Gathered A row for this lane's M index (same for both K-halves).
    const float* arow = vecs + (long)data[row0 + l16] * VEC_DIM;
    const int    n    = col0 + l16;

    v8f acc = {};                            // C starts at 0; bias in epilogue
    #pragma unroll 5
    for (int k = 0; k < VEC_DIM; k += 4) {   // 300 = 75 * 4, exact
        v2f a, b;
        a.x = arow[k + 2 * half + 0];
        a.y = arow[k + 2 * half + 1];
        b.x = W[(k + 2 * half + 0) * DIM + n];
        b.y = W[(k + 2 * half + 1) * DIM + n];
        acc = __builtin_amdgcn_wmma_f32_16x16x4_f32(
            /*neg_a=*/false, a, /*neg_b=*/false, b,
            /*c_mod=*/(short)0, acc, /*reuse_a=*/false, /*reuse_b=*/false);
    }

    const float bj = bias[n];
    #pragma unroll
    for (int r = 0; r < 8; ++r) {
        const int row = row0 + r + 8 * half;
        emb[row * DIM + n] = acc[r] + bj;
    }
}

// ---------------------------------------------------------------------------
// Kernel 2: the sequential chain (4095 dependent 128x128 matvecs + ReLU).
// Single workgroup, 128 threads (1 wave32 per SIMD of one WGP).  Thread j
// keeps W0[:,j] in 128 VGPRs.  Double-buffered activation vector in LDS ->
// ONE barrier per step.  4 independent accumulators break the FMA chain
// (enables v_dual_fmac / pipelining).  Next emb row is loaded into a
// register before the barrier so its latency overlaps barrier + dot.
// ---------------------------------------------------------------------------
__device__ __forceinline__ float dot128(const float* __restrict__ t,
                                        const float* __restrict__ w,
                                        float bias0)
{
    const float4* tv = (const float4*)t;
    float a0 = bias0, a1 = 0.0f, a2 = 0.0f, a3 = 0.0f;
    #pragma unroll
    for (int k4 = 0; k4 < DIM / 4; ++k4) {
        float4 x = tv[k4];
        a0 = fmaf(w[4 * k4 + 0], x.x, a0);
        a1 = fmaf(w[4 * k4 + 1], x.y, a1);
        a2 = fmaf(w[4 * k4 + 2], x.z, a2);
        a3 = fmaf(w[4 * k4 + 3], x.w, a3);
    }
    return (a0 + a1) + (a2 + a3);
}

__global__ __launch_bounds__(128, 1) void chain_kernel(
    const float* __restrict__ emb,    // [4096, 128]
    const float* __restrict__ eW,     // edge_weights, W0 = eW[0:128,0:128]
    const float* __restrict__ eB,     // edge_biases,  b0 = eB[0:128]
    float*       __restrict__ h_out)  // [128]
{
    __shared__ float t0[DIM];
    __shared__ float t1[DIM];

    const int j = threadIdx.x;        // 0..127

    // W0 column j -> 128 VGPRs (coalesced across lanes for each k).
    float w[DIM];
    #pragma unroll
    for (int k = 0; k < DIM; ++k) w[k] = eW[k * DIM + j];
    const float b0j = eB[j];

    // Step 0 (uses t0): e = b0 + relu(emb[0]) @ W0
    t0[j] = fmaxf(emb[j], 0.0f);
    float nxt = emb[DIM + j];                 // emb row 1, off critical path
    __syncthreads();
    float e = dot128(t0, w, b0j);

    // Steps 1..4094 (4094 steps = 2047 pairs; odd step -> t1, even -> t0).
    for (int i = 1; i < N_TOK - 1; i += 2) {
        // --- step i (odd): buffer t1 ---
        t1[j] = fmaxf((e + nxt) * 0.5f, 0.0f);
        nxt = emb[(i + 1) * DIM + j];         // issue before barrier
        __builtin_prefetch(&emb[(i + 16) * DIM + j], 0, 1);
        __syncthreads();
        e = dot128(t1, w, b0j);

        // --- step i+1 (even): buffer t0 ---
        t0[j] = fmaxf((e + nxt) * 0.5f, 0.0f);
        nxt = emb[(i + 2) * DIM + j];         // i+2 <= 4095: last read is row
                                              // 4095, used by the epilogue
        __syncthreads();
        e = dot128(t0, w, b0j);
    }

    // h = relu((emb[4095] + e) * 0.5); nxt already holds emb row 4095.
    h_out[j] = fmaxf((e + nxt) * 0.5f, 0.0f);
}

// ---------------------------------------------------------------------------
// Kernel 3: score[e] = h . (W_e @ s) + b_e . s   -- streams 131 MB, HBM/L2
// bound (floor ~5.6 us from HBM; L2-resident on graph replay).  One
// 256-thread block per edge; float4 streaming loads; LDS reduction.
// ---------------------------------------------------------------------------
__global__ __launch_bounds__(256) void score_kernel(
    const float* __restrict__ eW,     // [2000, 128, 128]
    const float* __restrict__ eB,     // [2000, 128]
    const float* __restrict__ s,      // [128] score weights
    const float* __restrict__ h,      // [128]
    float*       __restrict__ out)    // [2000]
{
    __shared__ float hs[DIM];
    __shared__ float ss[DIM];
    __shared__ float red[256];

    const int e = blockIdx.x;
    const int t = threadIdx.x;

    if (t < DIM) { hs[t] = h[t]; ss[t] = s[t]; }
    __syncthreads();

    const float4* Wv = (const float4*)(eW + (size_t)e * DIM * DIM);

    float acc = 0.0f;
    #pragma unroll
    for (int i = 0; i < 16; ++i) {                 // 16 * 256 = 4096 float4s
        const int f4 = t + i * 256;                // float4 index
        const int d  = (t >> 5) + 8 * i;           // row  (= (4*f4) >> 7)
        const int k  = (t << 2) & (DIM - 1);       // col within row
        const float4 wv = Wv[f4];
        acc += hs[d] * (wv.x * ss[k + 0] + wv.y * ss[k + 1] +
                        wv.z * ss[k + 2] + wv.w * ss[k + 3]);
    }
    if (t < DIM) acc += eB[(size_t)e * DIM + t] * ss[t];   // fold b_e . s

    red[t] = acc;
    __syncthreads();
    #pragma unroll
    for (int off = 128; off > 0; off >>= 1) {
        if (t < off) red[t] += red[t + off];
        __syncthreads();
    }
    if (t == 0) out[e] = red[0];
}

// ---------------------------------------------------------------------------
// Launcher.  ws layout: [0 .. 4096*128) = emb (f32), then [.. +128) = h.
// ---------------------------------------------------------------------------
extern "C" void kernel_launch(void* const* d_in, const int* in_sizes, int n_in,
                              void* d_out, int out_size, void* d_ws, size_t ws_size,
                              hipStream_t stream) {
    const int*   data = (const int*)  d_in[0];
    const float* vecs = (const float*)d_in[1];
    const float* W    = (const float*)d_in[2];
    const float* bias = (const float*)d_in[3];
    const float* eW   = (const float*)d_in[4];
    const float* eB   = (const float*)d_in[5];
    const float* sw   = (const float*)d_in[6];
    float*       out  = (float*)d_out;

    float* emb = (float*)d_ws;                 // 4096*128 f32 = 2 MB
    float* h   = emb + (size_t)N_TOK * DIM;    // 128 f32

    // 2048 tiles, 1 wave per tile, 4 waves per block -> 512 blocks.
    emb_gemm_wmma<<<512, 128, 0, stream>>>(data, vecs, W, bias, emb);
    chain_kernel<<<1, 128, 0, stream>>>(emb, eW, eB, h);
    score_kernel<<<EDGES, 256, 0, stream>>>(eW, eB, sw, h, out);
}